// GAT_19396072309041
// MI455X (gfx1250) — compile-verified
//
#include <hip/hip_runtime.h>
#include <cmath>

// GAT forward for MI455X (gfx1250, wave32).
// GEMMs via V_WMMA_F32_16X16X4_F32: one wave -> 16x64 tile (4 WMMA tiles share
// one A fragment), K stepped by 4, incremental pointers, bias handling
// compile-time templated. Edge softmax/aggregation via hw f32 atomics
// (L2-resident working set: feat 51MB + agg 51MB < 192MB L2).

#define N_NODES 50000
#define N_EDGES 800000
#define HEADS 4
#define DFEAT 64
#define HD 256        // HEADS * DFEAT
#define IN_F 128

typedef __attribute__((ext_vector_type(2))) float v2f;
typedef __attribute__((ext_vector_type(8))) float v8f;

__device__ __forceinline__ void atomicMaxF(float* addr, float val) {
  // monotone bit-pattern trick: works for mixed signs with -inf init
  if (val >= 0.0f) atomicMax((int*)addr, __float_as_int(val));
  else             atomicMin((unsigned int*)addr, __float_as_uint(val));
}

// ---------------------------------------------------------------------------
// C[M,Ncols] = (A[M,K] + abias[K]?) @ B[K,Ncols] + cbias[Ncols]?
// One wave per 16x64 output tile: 4 adjacent 16x16 WMMA tiles reuse the A frag.
// A frag (16x4): lane<16 holds M=lane,K={k,k+1}; lane>=16 holds M=lane-16,K={k+2,k+3}
// B frag (4x16): col=lane&15, VGPR v -> K = k + v + 2*(lane>=16)
// D (16x16):     VGPR v -> M = 16*mt + v + 8*(lane>=16), N = tile_base + (lane&15)
// ---------------------------------------------------------------------------
template <bool HAS_ABIAS, bool HAS_CBIAS>
__global__ void gemm_wmma_f32_t(const float* __restrict__ A,
                                const float* __restrict__ B,
                                const float* __restrict__ abias,
                                const float* __restrict__ cbias,
                                float* __restrict__ C,
                                int M, int K, int Ncols) {
  int wave = (int)((blockIdx.x * blockDim.x + threadIdx.x) >> 5);
  int lane = threadIdx.x & 31;
  int gtiles = Ncols >> 6;                 // 64-column groups
  int total = (M >> 4) * gtiles;
  if (wave >= total) return;               // wave-uniform: EXEC stays all-ones
  int mt = wave / gtiles;
  int gt = wave - mt * gtiles;
  int col = lane & 15;
  int koff = (lane >> 4) << 1;             // 0 or 2
  const float* Ap = A + (size_t)(mt * 16 + col) * (size_t)K + koff;
  const float* Bp = B + (size_t)koff * (size_t)Ncols + gt * 64 + col;
  const float* ab = abias + koff;          // only dereferenced if HAS_ABIAS
  v8f acc0 = {}, acc1 = {}, acc2 = {}, acc3 = {};
  for (int k = 0; k < K; k += 4) {
    v2f a;
    a.x = Ap[0];
    a.y = Ap[1];
    if (HAS_ABIAS) { a.x += ab[0]; a.y += ab[1]; ab += 4; }
    v2f b0, b1, b2, b3;
    b0.x = Bp[0];  b0.y = Bp[Ncols];
    b1.x = Bp[16]; b1.y = Bp[Ncols + 16];
    b2.x = Bp[32]; b2.y = Bp[Ncols + 32];
    b3.x = Bp[48]; b3.y = Bp[Ncols + 48];
    acc0 = __builtin_amdgcn_wmma_f32_16x16x4_f32(false, a, false, b0, (short)0, acc0, false, false);
    acc1 = __builtin_amdgcn_wmma_f32_16x16x4_f32(false, a, false, b1, (short)0, acc1, false, false);
    acc2 = __builtin_amdgcn_wmma_f32_16x16x4_f32(false, a, false, b2, (short)0, acc2, false, false);
    acc3 = __builtin_amdgcn_wmma_f32_16x16x4_f32(false, a, false, b3, (short)0, acc3, false, false);
    Ap += 4;
    Bp += (size_t)4 * (size_t)Ncols;
  }
  float cb0 = 0.f, cb1 = 0.f, cb2 = 0.f, cb3 = 0.f;
  if (HAS_CBIAS) {
    const float* cbp = cbias + gt * 64 + col;
    cb0 = cbp[0]; cb1 = cbp[16]; cb2 = cbp[32]; cb3 = cbp[48];
  }
  int mbase = mt * 16 + ((lane >> 4) << 3);
  float* Cp = C + (size_t)mbase * (size_t)Ncols + gt * 64 + col;
#pragma unroll
  for (int v = 0; v < 8; ++v) {
    Cp[0]  = acc0[v] + cb0;
    Cp[16] = acc1[v] + cb1;
    Cp[32] = acc2[v] + cb2;
    Cp[48] = acc3[v] + cb3;
    Cp += Ncols;
  }
}

// el[n,h] = <feat[n,h,:], al[h,:]>, er likewise
__global__ void el_er_kernel(const float* __restrict__ feat,
                             const float* __restrict__ al,
                             const float* __restrict__ ar,
                             float* __restrict__ el, float* __restrict__ er) {
  int i = blockIdx.x * blockDim.x + threadIdx.x;
  if (i >= N_NODES * HEADS) return;
  int n = i >> 2, h = i & 3;
  const float4* f = (const float4*)(feat + (size_t)n * HD + h * DFEAT);
  const float4* a = (const float4*)(al + h * DFEAT);
  const float4* b = (const float4*)(ar + h * DFEAT);
  float sl = 0.f, sr = 0.f;
#pragma unroll
  for (int d = 0; d < DFEAT / 4; ++d) {
    float4 v = f[d], av = a[d], bv = b[d];
    sl += v.x * av.x + v.y * av.y + v.z * av.z + v.w * av.w;
    sr += v.x * bv.x + v.y * bv.y + v.z * bv.z + v.w * bv.w;
  }
  el[i] = sl; er[i] = sr;
}

__global__ void fill_kernel(float* __restrict__ p, float v, int n) {
  int i = blockIdx.x * blockDim.x + threadIdx.x;
  if (i < n) p[i] = v;
}

// pass A: e = leaky_relu(el[src]+er[dst]); segment max via hw atomic
__global__ void edge_score_max(const int* __restrict__ src, const int* __restrict__ dst,
                               const float* __restrict__ el, const float* __restrict__ er,
                               float* __restrict__ ebuf, float* __restrict__ emax) {
  int i = blockIdx.x * blockDim.x + threadIdx.x;
  if (i >= N_EDGES * HEADS) return;
  int e = i >> 2, h = i & 3;
  float x = el[src[e] * HEADS + h] + er[dst[e] * HEADS + h];
  x = (x > 0.f) ? x : 0.2f * x;
  ebuf[i] = x;
  atomicMaxF(&emax[dst[e] * HEADS + h], x);
}

// pass B: a = exp(e - emax[dst]); segment sum
__global__ void edge_exp_sum(const int* __restrict__ dst,
                             const float* __restrict__ emax,
                             float* __restrict__ ebuf, float* __restrict__ esum) {
  int i = blockIdx.x * blockDim.x + threadIdx.x;
  if (i >= N_EDGES * HEADS) return;
  int e = i >> 2, h = i & 3;
  int dh = dst[e] * HEADS + h;
  float a = expf(ebuf[i] - emax[dh]);
  ebuf[i] = a;
  unsafeAtomicAdd(&esum[dh], a);
}

// pass C: alpha = a / ssum[dst]
__global__ void edge_alpha(const int* __restrict__ dst,
                           const float* __restrict__ esum, float* __restrict__ ebuf) {
  int i = blockIdx.x * blockDim.x + threadIdx.x;
  if (i >= N_EDGES * HEADS) return;
  int e = i >> 2, h = i & 3;
  ebuf[i] = ebuf[i] / esum[dst[e] * HEADS + h];
}

// agg[dst] += feat[src] * alpha   (bandwidth-critical; float4 gathers,
// L2-resident operands, hw global_atomic_add_f32 scatter)
__global__ void edge_aggregate(const int* __restrict__ src, const int* __restrict__ dst,
                               const float* __restrict__ feat, const float* __restrict__ alpha,
                               float* __restrict__ agg) {
  int gid = blockIdx.x * blockDim.x + threadIdx.x;
  if (gid >= N_EDGES * (HD / 4)) return;
  int e = gid >> 6, c4 = gid & 63, h = c4 >> 4;
  float al_ = alpha[e * HEADS + h];
  const float4* f4 = (const float4*)(feat + (size_t)src[e] * HD);
  float4 v = f4[c4];
  float* o = agg + (size_t)dst[e] * HD + c4 * 4;
  unsafeAtomicAdd(o + 0, v.x * al_);
  unsafeAtomicAdd(o + 1, v.y * al_);
  unsafeAtomicAdd(o + 2, v.z * al_);
  unsafeAtomicAdd(o + 3, v.w * al_);
}

// hout[n,d] = mean_h(agg[n,h,d] + bias[h,d])
__global__ void head_mean_kernel(const float* __restrict__ agg,
                                 const float* __restrict__ bias,
                                 float* __restrict__ hout) {
  int i = blockIdx.x * blockDim.x + threadIdx.x;
  if (i >= N_NODES * DFEAT) return;
  int n = i >> 6, d = i & 63;
  const float* a = agg + (size_t)n * HD + d;
  float s = (a[0] + bias[d]) + (a[64] + bias[64 + d]) +
            (a[128] + bias[128 + d]) + (a[192] + bias[192 + d]);
  hout[i] = 0.25f * s;
}

// LayerNorm over rows of 64, one wave32 per row, shuffle reductions
__global__ void layernorm_kernel(const float* __restrict__ x, float* __restrict__ y) {
  int wave = (int)((blockIdx.x * blockDim.x + threadIdx.x) >> 5);
  int lane = threadIdx.x & 31;
  if (wave >= N_NODES) return;
  const float* r = x + (size_t)wave * DFEAT;
  float v0 = r[lane], v1 = r[lane + 32];
  float s = v0 + v1;
#pragma unroll
  for (int m = 16; m > 0; m >>= 1) s += __shfl_xor(s, m, 32);
  float mu = s * (1.0f / 64.0f);
  float d0 = v0 - mu, d1 = v1 - mu;
  float q = d0 * d0 + d1 * d1;
#pragma unroll
  for (int m = 16; m > 0; m >>= 1) q += __shfl_xor(q, m, 32);
  float inv = rsqrtf(q * (1.0f / 64.0f) + 1e-5f);
  float* yr = y + (size_t)wave * DFEAT;
  yr[lane] = d0 * inv;
  yr[lane + 32] = d1 * inv;
}

// ---------------------------------------------------------------------------
static void launch_gemm(const float* A, const float* Bm, float* C,
                        int M, int K, int Ncols, hipStream_t stream) {
  int waves = (M >> 4) * (Ncols >> 6);
  int threads = waves * 32;
  gemm_wmma_f32_t<false, false><<<(threads + 255) / 256, 256, 0, stream>>>(
      A, Bm, nullptr, nullptr, C, M, K, Ncols);
}

static void launch_gemm_bias(const float* A, const float* Bm, const float* abias,
                             const float* cbias, float* C, int M, int K, int Ncols,
                             hipStream_t stream) {
  int waves = (M >> 4) * (Ncols >> 6);
  int threads = waves * 32;
  gemm_wmma_f32_t<true, true><<<(threads + 255) / 256, 256, 0, stream>>>(
      A, Bm, abias, cbias, C, M, K, Ncols);
}

static void run_edge_phase(const int* src, const int* dst, const float* feat,
                           const float* al, const float* ar,
                           float* el, float* er, float* emax, float* esum,
                           float* ebuf, float* agg, hipStream_t stream) {
  const int B = 256;
  const int NH = N_NODES * HEADS;
  const int EH = N_EDGES * HEADS;
  const int EW = N_EDGES * (HD / 4);
  el_er_kernel<<<(NH + B - 1) / B, B, 0, stream>>>(feat, al, ar, el, er);
  fill_kernel<<<(NH + B - 1) / B, B, 0, stream>>>(emax, -__builtin_inff(), NH);
  fill_kernel<<<(NH + B - 1) / B, B, 0, stream>>>(esum, 0.0f, NH);
  fill_kernel<<<(N_NODES * HD + B - 1) / B, B, 0, stream>>>(agg, 0.0f, N_NODES * HD);
  edge_score_max<<<(EH + B - 1) / B, B, 0, stream>>>(src, dst, el, er, ebuf, emax);
  edge_exp_sum<<<(EH + B - 1) / B, B, 0, stream>>>(dst, emax, ebuf, esum);
  edge_alpha<<<(EH + B - 1) / B, B, 0, stream>>>(dst, esum, ebuf);
  edge_aggregate<<<(EW + B - 1) / B, B, 0, stream>>>(src, dst, feat, ebuf, agg);
}

extern "C" void kernel_launch(void* const* d_in, const int* in_sizes, int n_in,
                              void* d_out, int out_size, void* d_ws, size_t ws_size,
                              hipStream_t stream) {
  const float* in_feat = (const float*)d_in[0];
  const int*   src     = (const int*)d_in[1];
  const int*   dst     = (const int*)d_in[2];
  const float* W1      = (const float*)d_in[3];
  const float* al1     = (const float*)d_in[4];
  const float* ar1     = (const float*)d_in[5];
  const float* b1      = (const float*)d_in[6];
  const float* Wh      = (const float*)d_in[7];   // [3, 64, 256]
  const float* alh     = (const float*)d_in[8];   // [3, 4, 64]
  const float* arh     = (const float*)d_in[9];   // [3, 4, 64]
  const float* bh      = (const float*)d_in[10];  // [3, 256]
  const float* Wo      = (const float*)d_in[11];  // [256, 64]
  const float* bo      = (const float*)d_in[12];  // [64]
  float* out = (float*)d_out;

  float* ws   = (float*)d_ws;
  float* feat = ws;  ws += (size_t)N_NODES * HD;     // 12.8M floats
  float* agg  = ws;  ws += (size_t)N_NODES * HD;     // 12.8M
  float* hbuf = ws;  ws += (size_t)N_NODES * DFEAT;  // 3.2M
  float* el   = ws;  ws += (size_t)N_NODES * HEADS;
  float* er   = ws;  ws += (size_t)N_NODES * HEADS;
  float* emax = ws;  ws += (size_t)N_NODES * HEADS;
  float* esum = ws;  ws += (size_t)N_NODES * HEADS;
  float* ebuf = ws;  ws += (size_t)N_EDGES * HEADS;  // 3.2M

  const int B = 256;

  // ---- conv1: IN=128 -> [N,H,D], head-mean -> hbuf[N,64]
  launch_gemm(in_feat, W1, feat, N_NODES, IN_F, HD, stream);
  run_edge_phase(src, dst, feat, al1, ar1, el, er, emax, esum, ebuf, agg, stream);
  head_mean_kernel<<<(N_NODES * DFEAT + B - 1) / B, B, 0, stream>>>(agg, b1, hbuf);

  // ---- hidden convs 2..3 (mean over heads)
  for (int i = 0; i < 2; ++i) {
    launch_gemm(hbuf, Wh + (size_t)i * DFEAT * HD, feat, N_NODES, DFEAT, HD, stream);
    run_edge_phase(src, dst, feat, alh + i * HD, arh + i * HD,
                   el, er, emax, esum, ebuf, agg, stream);
    head_mean_kernel<<<(N_NODES * DFEAT + B - 1) / B, B, 0, stream>>>(
        agg, bh + i * HD, hbuf);
  }

  // ---- conv4 (keep [N, H*D]; its bias fused into out-projection's A operand)
  launch_gemm(hbuf, Wh + (size_t)2 * DFEAT * HD, feat, N_NODES, DFEAT, HD, stream);
  run_edge_phase(src, dst, feat, alh + 2 * HD, arh + 2 * HD,
                 el, er, emax, esum, ebuf, agg, stream);

  // ---- out_ln: (agg + bh[2]) @ Wo + bo -> hbuf[N,64]
  launch_gemm_bias(agg, Wo, bh + 2 * HD, bo, hbuf, N_NODES, HD, DFEAT, stream);

  // ---- LayerNorm (no affine) -> d_out
  layernorm_kernel<<<(N_NODES * 32 + B - 1) / B, B, 0, stream>>>(hbuf, out);
}